// LJ_repulsive_68367289417769
// MI455X (gfx1250) — compile-verified
//
#include <hip/hip_runtime.h>
#include <hip/hip_bf16.h>

#define TILE_I 16
#define TILE_J 256
#define NTHREADS 256

#define SIGMA_RAW   3.405f
#define EPSILON_RAW 0.0103f
#define CUTOFF2     25.0f

#if defined(__has_builtin)
#  if __has_builtin(__builtin_amdgcn_global_load_async_to_lds_b128)
#    define HAVE_ASYNC_LDS 1
#  endif
#endif
#ifndef HAVE_ASYNC_LDS
#  define HAVE_ASYNC_LDS 0
#endif

// Builtin signature (from clang diagnostic): src is AS1 pointer to 4xi32 vector.
typedef int v4i __attribute__((vector_size(16)));
typedef __attribute__((address_space(1))) v4i gv4i;   // global
typedef __attribute__((address_space(3))) v4i lv4i;   // LDS

__device__ __forceinline__ void wait_asynccnt_zero() {
#if defined(__has_builtin) && __has_builtin(__builtin_amdgcn_s_wait_asynccnt)
    __builtin_amdgcn_s_wait_asynccnt(0);
#else
    asm volatile("s_wait_asynccnt 0" ::: "memory");
#endif
}

__global__ __launch_bounds__(NTHREADS)
void lj_repulsive_kernel(const float* __restrict__ q,
                         const float* __restrict__ cell,
                         float* __restrict__ out, int N)
{
    __shared__ float qj[TILE_J * 3];   // j-tile coords, AoS, 3072 B
    __shared__ float qi[TILE_I * 3];   // i-tile coords, AoS, 192 B

    const int tid = threadIdx.x;
    const int i0  = blockIdx.y * TILE_I;
    const int j0  = blockIdx.x * TILE_J;
    const int it  = tid >> 6;          // 0..3  : which group of 4 rows
    const int jq  = tid & 63;          // 0..63 : which group of 4 cols
    const int ib  = it * 4;
    const int jb  = jq * 4;

    // ---- fast path: tile entirely at/below diagonal -> all zeros ----
    if (i0 >= j0 + TILE_J - 1) {
        const float4 z = make_float4(0.f, 0.f, 0.f, 0.f);
        #pragma unroll
        for (int r = 0; r < 4; ++r)
            *(float4*)(out + (size_t)(i0 + ib + r) * N + (j0 + jb)) = z;
        return;
    }

    // ---- stage coordinate tiles into LDS (CDNA5 async copy path) ----
#if HAVE_ASYNC_LDS
    if (tid < (TILE_J * 3) / 4) {                 // 192 lanes x b128
        __builtin_amdgcn_global_load_async_to_lds_b128(
            (gv4i*)(q + (size_t)j0 * 3 + tid * 4),
            (lv4i*)(qj + tid * 4), 0, 0);
    } else if (tid < (TILE_J * 3) / 4 + (TILE_I * 3) / 4) {   // 12 lanes x b128
        const int t = tid - (TILE_J * 3) / 4;
        __builtin_amdgcn_global_load_async_to_lds_b128(
            (gv4i*)(q + (size_t)i0 * 3 + t * 4),
            (lv4i*)(qi + t * 4), 0, 0);
    }
    wait_asynccnt_zero();
#else
    for (int t = tid; t < TILE_J * 3; t += NTHREADS) qj[t] = q[(size_t)j0 * 3 + t];
    if (tid < TILE_I * 3) qi[tid] = q[(size_t)i0 * 3 + tid];
#endif
    __syncthreads();

    // ---- uniform constants ----
    const float Lx = cell[0], Ly = cell[1], Lz = cell[2];
    const float iLx = 1.0f / Lx, iLy = 1.0f / Ly, iLz = 1.0f / Lz;
    // 4 * exp(eps) * (exp(sigma))^12 = 4 * exp(eps + 12*sigma)
    const float C = 4.0f * __expf(EPSILON_RAW + 12.0f * SIGMA_RAW);

    // ---- pull micro-tile coords into registers ----
    float xi[4], yi[4], zi[4], xj[4], yj[4], zj[4];
    #pragma unroll
    for (int r = 0; r < 4; ++r) {
        xi[r] = qi[(ib + r) * 3 + 0];
        yi[r] = qi[(ib + r) * 3 + 1];
        zi[r] = qi[(ib + r) * 3 + 2];
    }
    #pragma unroll
    for (int c = 0; c < 4; ++c) {
        xj[c] = qj[(jb + c) * 3 + 0];
        yj[c] = qj[(jb + c) * 3 + 1];
        zj[c] = qj[(jb + c) * 3 + 2];
    }

    // ---- 4x4 micro-tile: min-image LJ repulsive ----
    #pragma unroll
    for (int r = 0; r < 4; ++r) {
        const int i = i0 + ib + r;
        float res[4];
        #pragma unroll
        for (int c = 0; c < 4; ++c) {
            const int j = j0 + jb + c;
            float dx = xi[r] - xj[c];
            float dy = yi[r] - yj[c];
            float dz = zi[r] - zj[c];
            dx = fmaf(-Lx, rintf(dx * iLx), dx);   // v_rndne_f32 (RNE, matches jnp.round)
            dy = fmaf(-Ly, rintf(dy * iLy), dy);
            dz = fmaf(-Lz, rintf(dz * iLz), dz);
            const float d2  = fmaf(dx, dx, fmaf(dy, dy, dz * dz));
            const float inv = __builtin_amdgcn_rcpf(d2);     // v_rcp_f32
            const float p2  = inv * inv;
            const float p6  = p2 * p2 * p2;
            const bool  m   = (i < j) && (d2 < CUTOFF2);
            res[c] = m ? (C * p6) : 0.0f;
        }
        *(float4*)(out + (size_t)i * N + (j0 + jb)) =
            make_float4(res[0], res[1], res[2], res[3]);
    }
}

extern "C" void kernel_launch(void* const* d_in, const int* in_sizes, int n_in,
                              void* d_out, int out_size, void* d_ws, size_t ws_size,
                              hipStream_t stream) {
    const float* q    = (const float*)d_in[0];   // [N,3] fp32
    const float* cell = (const float*)d_in[1];   // [3]   fp32
    float* out = (float*)d_out;                  // [N,N] fp32
    const int N = in_sizes[0] / 3;               // 4096

    dim3 grid(N / TILE_J, N / TILE_I);           // (16, 256)
    lj_repulsive_kernel<<<grid, NTHREADS, 0, stream>>>(q, cell, out, N);
}